// VanillaGNNModel_28020366639510
// MI455X (gfx1250) — compile-verified
//
#include <hip/hip_runtime.h>

// ---------------------------------------------------------------------------
// Problem constants (match the JAX reference)
// ---------------------------------------------------------------------------
#define N_    10000                // op nodes
#define NC_   1000                 // nconfig nodes
#define C_    32                   // configs per graph
#define FOP_  140                  // op feature dim
#define FCFG_ 18                   // per-node config feature dim
#define E_    40000                // feed edges
#define G_    8                    // graphs
#define EMB_  32                   // opcode embedding dim
#define H_    32                   // hidden dim
#define DOP_  (FOP_ + EMB_)        // 172  (op_feats ++ emb)
#define D0_   (FCFG_ + DOP_)       // 190  prenet input dim
#define D0P_  192                  // padded to multiple of 4 (WMMA K)
#define DG_   (FCFG_ + H_)         // 50   GC input dim
#define DGP_  52                   // padded
#define CH_   (C_ * H_)            // 1024 floats per node of x

typedef float v2f __attribute__((ext_vector_type(2)));
typedef float v8f __attribute__((ext_vector_type(8)));

#if defined(__gfx1250__) && __has_builtin(__builtin_amdgcn_wmma_f32_16x16x4_f32)
#define USE_WMMA_F32 1
#else
#define USE_WMMA_F32 0
#endif

__device__ __forceinline__ float lrelu(float v) { return v >= 0.f ? v : 0.2f * v; }

// ---------------------------------------------------------------------------
// 16x16 output tile of  X[32 x DinPad] @ Wp[DinPad x 32]  for one wave32.
// X lives in LDS (row stride XS, zero-padded columns); Wp is a global,
// row-major, ZERO-PADDED weight matrix (DinPad rows) so the K-loop carries
// no bounds checks at all: pure ds_load + global_load + v_wmma.
// Result uses the WMMA f32 C/D layout:
//   acc[v] -> (row = rowBase + v + 8*(lane>=16), col = colBase + lane%16)
// ---------------------------------------------------------------------------
template <int DinPad, int XS>
__device__ __forceinline__ v8f tile_mm(const float* __restrict__ X,
                                       const float* __restrict__ Wp,
                                       int rowBase, int colBase, int lane) {
  v8f acc = {};
  const int half = lane >> 4;   // 0 or 1
  const int lr   = lane & 15;
#if USE_WMMA_F32
  // A 16x4 f32 layout: lane lr holds row (rowBase+lr), VGPR j holds K = 2*half + j
  // B 4x16  f32 layout: lane lr holds col (colBase+lr), VGPR j holds K = 2*half + j
  const float* xrow = X + (rowBase + lr) * XS;
  const float* wcol = Wp + colBase + lr;
#pragma unroll 4
  for (int kb = 0; kb < DinPad; kb += 4) {
    const int k0 = kb + 2 * half;
    v2f a, b;
    a.x = xrow[k0];
    a.y = xrow[k0 + 1];
    b.x = wcol[k0 * H_];
    b.y = wcol[(k0 + 1) * H_];
    acc = __builtin_amdgcn_wmma_f32_16x16x4_f32(false, a, false, b,
                                                (short)0, acc, false, false);
  }
#else
  for (int k = 0; k < DinPad; ++k) {
    const float w = Wp[k * H_ + colBase + lr];
#pragma unroll
    for (int v = 0; v < 8; ++v)
      acc[v] += X[(rowBase + v + 8 * half) * XS + k] * w;
  }
#endif
  return acc;
}

// ---------------------------------------------------------------------------
// Elementwise / scatter kernels
// ---------------------------------------------------------------------------
__global__ __launch_bounds__(256) void k_fill(float* __restrict__ p, long long n, float v) {
  long long i = (long long)blockIdx.x * 256 + threadIdx.x;
  if (i < n) p[i] = v;
}

// Zero-pad a row-major [Din x 32] weight matrix to [DinPad x 32].
__global__ __launch_bounds__(256) void k_pad_w(const float* __restrict__ in,
                                               float* __restrict__ out,
                                               int Din, int DinPad) {
  int i = blockIdx.x * 256 + threadIdx.x;
  if (i >= DinPad * H_) return;
  const int k = i / H_;
  out[i] = (k < Din) ? in[i] : 0.f;
}

// op_full[n] = concat(op_feats[n], emb[op_codes[n]])
__global__ __launch_bounds__(256) void k_op_full(const float* __restrict__ op_feats,
                                                 const int* __restrict__ op_codes,
                                                 const float* __restrict__ emb,
                                                 float* __restrict__ out) {
  long long i = (long long)blockIdx.x * 256 + threadIdx.x;
  if (i >= (long long)N_ * DOP_) return;
  const int n = (int)(i / DOP_), d = (int)(i % DOP_);
  out[i] = (d < FOP_) ? op_feats[(size_t)n * FOP_ + d]
                      : emb[(size_t)op_codes[n] * EMB_ + (d - FOP_)];
}

// deg init to 1 beforehand; each edge adds 1 to src and tgt
__global__ __launch_bounds__(256) void k_deg(const int* __restrict__ fsrc,
                                             const int* __restrict__ ftgt,
                                             float* __restrict__ deg) {
  int e = blockIdx.x * 256 + threadIdx.x;
  if (e >= E_) return;
  atomicAdd(&deg[ftgt[e]], 1.f);
  atomicAdd(&deg[fsrc[e]], 1.f);
}

__global__ __launch_bounds__(256) void k_rsqrt(float* __restrict__ d, int n) {
  int i = blockIdx.x * 256 + threadIdx.x;
  if (i < n) d[i] = rsqrtf(d[i]);
}

// cfg_scaled[ctgt[i]] += 100 * config_feats[csrc[i]]   (zeroed beforehand)
__global__ __launch_bounds__(256) void k_cfg_scatter(const float* __restrict__ cf,
                                                     const int* __restrict__ csrc,
                                                     const int* __restrict__ ctgt,
                                                     float* __restrict__ cfgs) {
  long long idx = (long long)blockIdx.x * 256 + threadIdx.x;
  if (idx >= (long long)NC_ * C_ * FCFG_) return;
  const int i = (int)(idx / (C_ * FCFG_));
  const int r = (int)(idx % (C_ * FCFG_));
  atomicAdd(&cfgs[(size_t)ctgt[i] * (C_ * FCFG_) + r],
            100.f * cf[(size_t)csrc[i] * (C_ * FCFG_) + r]);
}

// y(m,c,f) = concat(cfg_scaled, x)(m,c,f) * dis[m]   (original y, pre-scatter)
__device__ __forceinline__ float y_val(const float* __restrict__ cfgs,
                                       const float* __restrict__ x,
                                       const float* __restrict__ dis,
                                       int m, int c, int f) {
  float v = (f < FCFG_) ? cfgs[((size_t)m * C_ + c) * FCFG_ + f]
                        : x[((size_t)m * C_ + c) * H_ + (f - FCFG_)];
  return v * dis[m];
}

// agg = y (the I term of I + A + A^T)
__global__ __launch_bounds__(256) void k_agg_init(const float* __restrict__ cfgs,
                                                  const float* __restrict__ x,
                                                  const float* __restrict__ dis,
                                                  float* __restrict__ agg) {
  long long idx = (long long)blockIdx.x * 256 + threadIdx.x;
  if (idx >= (long long)N_ * C_ * DG_) return;
  const int r = (int)(idx % (C_ * DG_));
  const int n = (int)(idx / (C_ * DG_));
  const int c = r / DG_, f = r - c * DG_;
  agg[idx] = y_val(cfgs, x, dis, n, c, f);
}

// agg[tgt] += y[src]; agg[src] += y[tgt]   (original y values, functional semantics)
__global__ __launch_bounds__(256) void k_edge_scatter(const int* __restrict__ fsrc,
                                                      const int* __restrict__ ftgt,
                                                      const float* __restrict__ cfgs,
                                                      const float* __restrict__ x,
                                                      const float* __restrict__ dis,
                                                      float* __restrict__ agg) {
  long long idx = (long long)blockIdx.x * 256 + threadIdx.x;
  if (idx >= (long long)E_ * C_ * DG_) return;
  const int r = (int)(idx % (C_ * DG_));
  const int e = (int)(idx / (C_ * DG_));
  const int c = r / DG_, f = r - c * DG_;
  const int s = fsrc[e], t = ftgt[e];
  const float ys = y_val(cfgs, x, dis, s, c, f);
  const float yt = y_val(cfgs, x, dis, t, c, f);
  atomicAdd(&agg[(size_t)t * (C_ * DG_) + r], ys);
  atomicAdd(&agg[(size_t)s * (C_ * DG_) + r], yt);
}

// ---------------------------------------------------------------------------
// Prenet: x[n] = lrelu( mlp2( [cfg_scaled[n] | op_full[n]] ) )   [32 x 32]
// One block (4 waves) per node; each wave owns a 16x16 WMMA tile.
// ---------------------------------------------------------------------------
#define XS0_ 196   // LDS row stride for 192-wide input (bank-skewed)
#define HS_  34    // LDS row stride for 32-wide hidden

__global__ __launch_bounds__(128) void k_prenet(const float* __restrict__ cfgs,
                                                const float* __restrict__ opfull,
                                                const float* __restrict__ W0p,
                                                const float* __restrict__ b0,
                                                const float* __restrict__ W1,
                                                const float* __restrict__ b1,
                                                float* __restrict__ xout) {
  __shared__ float Xin[C_ * XS0_];
  __shared__ float Hmid[C_ * HS_];
  const int n = blockIdx.x, tid = threadIdx.x;
  for (int i = tid; i < C_ * D0P_; i += 128) {
    const int r = i / D0P_, k = i - r * D0P_;
    float v;
    if (k < FCFG_)    v = cfgs[((size_t)n * C_ + r) * FCFG_ + k];
    else if (k < D0_) v = opfull[(size_t)n * DOP_ + (k - FCFG_)];
    else              v = 0.f;
    Xin[r * XS0_ + k] = v;
  }
  __syncthreads();
  const int lane = tid & 31, wid = tid >> 5;
  const int rowBase = (wid >> 1) << 4, colBase = (wid & 1) << 4;
  const int lr = lane & 15, half = lane >> 4;
  const int col = colBase + lr;

  v8f acc = tile_mm<D0P_, XS0_>(Xin, W0p, rowBase, colBase, lane);
  const float bb0 = b0[col];
#pragma unroll
  for (int v = 0; v < 8; ++v)
    Hmid[(rowBase + v + 8 * half) * HS_ + col] = lrelu(acc[v] + bb0);
  __syncthreads();

  v8f acc2 = tile_mm<H_, HS_>(Hmid, W1, rowBase, colBase, lane);
  const float bb1 = b1[col];
  float* xo = xout + (size_t)n * CH_;
#pragma unroll
  for (int v = 0; v < 8; ++v)
    xo[(rowBase + v + 8 * half) * H_ + col] = lrelu(acc2[v] + bb1);
}

// ---------------------------------------------------------------------------
// GC layer MLP: x[n] += lrelu( mlp2( agg[n] * dis[n] ) )
// ---------------------------------------------------------------------------
#define XSG_ 56

__global__ __launch_bounds__(128) void k_gc_mlp(const float* __restrict__ agg,
                                                const float* __restrict__ dis,
                                                const float* __restrict__ W0p,
                                                const float* __restrict__ b0,
                                                const float* __restrict__ W1,
                                                const float* __restrict__ b1,
                                                float* __restrict__ x) {
  __shared__ float Xin[C_ * XSG_];
  __shared__ float Hmid[C_ * HS_];
  const int n = blockIdx.x, tid = threadIdx.x;
  const float ds = dis[n];
  for (int i = tid; i < C_ * DGP_; i += 128) {
    const int r = i / DGP_, k = i - r * DGP_;
    Xin[r * XSG_ + k] = (k < DG_) ? agg[((size_t)n * C_ + r) * DG_ + k] * ds : 0.f;
  }
  __syncthreads();
  const int lane = tid & 31, wid = tid >> 5;
  const int rowBase = (wid >> 1) << 4, colBase = (wid & 1) << 4;
  const int lr = lane & 15, half = lane >> 4;
  const int col = colBase + lr;

  v8f acc = tile_mm<DGP_, XSG_>(Xin, W0p, rowBase, colBase, lane);
  const float bb0 = b0[col];
#pragma unroll
  for (int v = 0; v < 8; ++v)
    Hmid[(rowBase + v + 8 * half) * HS_ + col] = lrelu(acc[v] + bb0);
  __syncthreads();

  v8f acc2 = tile_mm<H_, HS_>(Hmid, W1, rowBase, colBase, lane);
  const float bb1 = b1[col];
  float* xo = x + (size_t)n * CH_;
#pragma unroll
  for (int v = 0; v < 8; ++v) {
    const size_t o = (size_t)(rowBase + v + 8 * half) * H_ + col;
    xo[o] = xo[o] + lrelu(acc2[v] + bb1);   // residual
  }
}

// ---------------------------------------------------------------------------
// Merge + pooling + postnet
// ---------------------------------------------------------------------------
__global__ __launch_bounds__(256) void k_select(const int* __restrict__ sel,
                                                const float* __restrict__ xbp,
                                                float* __restrict__ xfull) {
  long long i = (long long)blockIdx.x * 256 + threadIdx.x;
  if (i >= (long long)N_ * CH_) return;
  if (sel[(int)(i / CH_)] != 0) xfull[i] = xbp[i];
}

// cfg_nodes[csrc[i]] += x[ctgt[i]]   (zeroed beforehand)
__global__ __launch_bounds__(256) void k_cfg_nodes(const int* __restrict__ csrc,
                                                   const int* __restrict__ ctgt,
                                                   const float* __restrict__ x,
                                                   float* __restrict__ cfg_nodes) {
  long long idx = (long long)blockIdx.x * 256 + threadIdx.x;
  if (idx >= (long long)NC_ * CH_) return;
  const int i = (int)(idx / CH_), r = (int)(idx % CH_);
  atomicAdd(&cfg_nodes[(size_t)csrc[i] * CH_ + r], x[(size_t)ctgt[i] * CH_ + r]);
}

__global__ __launch_bounds__(256) void k_nops(const int* __restrict__ gid,
                                              float* __restrict__ nops) {
  int n = blockIdx.x * 256 + threadIdx.x;
  if (n < N_) atomicAdd(&nops[gid[n]], 1.f);
}

__global__ __launch_bounds__(256) void k_graph_sum(const float* __restrict__ v,
                                                   const int* __restrict__ gid,
                                                   float* __restrict__ out,
                                                   long long total) {
  long long idx = (long long)blockIdx.x * 256 + threadIdx.x;
  if (idx >= total) return;
  const int node = (int)(idx / CH_), r = (int)(idx % CH_);
  atomicAdd(&out[(size_t)gid[node] * CH_ + r], v[idx]);
}

// One wave32 per (g, c): pooled = [mean_op | l2n(sum_op) | l2n(sum_cfg)] (96)
// out[g,c] = (lrelu(pooled @ W0) @ W1)
__global__ __launch_bounds__(32) void k_post(const float* __restrict__ sum_op,
                                             const float* __restrict__ sum_cfg,
                                             const float* __restrict__ nops,
                                             const float* __restrict__ W0,
                                             const float* __restrict__ W1,
                                             float* __restrict__ out) {
  const int g = blockIdx.x >> 5, c = blockIdx.x & 31, t = threadIdx.x;
  __shared__ float so[H_], sc[H_];
  const float a = sum_op[((size_t)g * C_ + c) * H_ + t];
  const float b = sum_cfg[((size_t)g * C_ + c) * H_ + t];
  so[t] = a; sc[t] = b;
  float ssa = a * a, ssb = b * b;
  for (int off = 16; off > 0; off >>= 1) {
    ssa += __shfl_xor(ssa, off, 32);
    ssb += __shfl_xor(ssb, off, 32);
  }
  const float inva  = rsqrtf(fmaxf(ssa, 1e-12f));
  const float invb  = rsqrtf(fmaxf(ssb, 1e-12f));
  const float inv_n = 1.f / fmaxf(nops[g], 1.f);
  __syncthreads();
  float acc = 0.f;
  for (int d = 0; d < H_; ++d) acc += so[d] * inv_n * W0[d * H_ + t];
  for (int d = 0; d < H_; ++d) acc += so[d] * inva  * W0[(H_ + d) * H_ + t];
  for (int d = 0; d < H_; ++d) acc += sc[d] * invb  * W0[(2 * H_ + d) * H_ + t];
  float o = lrelu(acc) * W1[t];
  for (int off = 16; off > 0; off >>= 1) o += __shfl_xor(o, off, 32);
  if (t == 0) out[(size_t)g * C_ + c] = o;
}

// ---------------------------------------------------------------------------
// Host orchestration
// ---------------------------------------------------------------------------
static inline unsigned blk(long long n) { return (unsigned)((n + 255) / 256); }

static void run_node_forward(const int* fsrc, const int* ftgt,
                             const int* csrc, const int* ctgt,
                             const float* cfg_feats, const float* op_full,
                             float* dis, float* cfgs, float* agg,
                             const float* pW0p, const float* pb0,
                             const float* pW1, const float* pb1,
                             const float* const gW0p[2], const float* const gb0[2],
                             const float* const gW1[2], const float* const gb1[2],
                             float* x_out, hipStream_t st) {
  k_fill<<<blk(N_), 256, 0, st>>>(dis, N_, 1.f);                       // deg = 1
  k_deg<<<blk(E_), 256, 0, st>>>(fsrc, ftgt, dis);                     // + A + A^T
  k_rsqrt<<<blk(N_), 256, 0, st>>>(dis, N_);                           // D^-1/2
  k_fill<<<blk((long long)N_ * C_ * FCFG_), 256, 0, st>>>(cfgs, (long long)N_ * C_ * FCFG_, 0.f);
  k_cfg_scatter<<<blk((long long)NC_ * C_ * FCFG_), 256, 0, st>>>(cfg_feats, csrc, ctgt, cfgs);
  k_prenet<<<N_, 128, 0, st>>>(cfgs, op_full, pW0p, pb0, pW1, pb1, x_out);
  for (int l = 0; l < 2; ++l) {
    k_agg_init<<<blk((long long)N_ * C_ * DG_), 256, 0, st>>>(cfgs, x_out, dis, agg);
    k_edge_scatter<<<blk((long long)E_ * C_ * DG_), 256, 0, st>>>(fsrc, ftgt, cfgs, x_out, dis, agg);
    k_gc_mlp<<<N_, 128, 0, st>>>(agg, dis, gW0p[l], gb0[l], gW1[l], gb1[l], x_out);
  }
}

extern "C" void kernel_launch(void* const* d_in, const int* in_sizes, int n_in,
                              void* d_out, int out_size, void* d_ws, size_t ws_size,
                              hipStream_t stream) {
  (void)in_sizes; (void)n_in; (void)out_size; (void)ws_size;
  // ---- inputs (setup_inputs order) ----
  const float* op_feats     = (const float*)d_in[0];
  const int*   op_codes     = (const int*)  d_in[1];
  const int*   selected     = (const int*)  d_in[2];
  const float* config_feats = (const float*)d_in[3];
  const int*   feed_src     = (const int*)  d_in[4];
  const int*   feed_tgt     = (const int*)  d_in[5];
  const int*   config_src   = (const int*)  d_in[6];
  const int*   config_tgt   = (const int*)  d_in[7];
  const int*   s_feed_src   = (const int*)  d_in[8];
  const int*   s_feed_tgt   = (const int*)  d_in[9];
  const int*   s_config_src = (const int*)  d_in[10];
  const int*   s_config_tgt = (const int*)  d_in[11];
  const int*   op_graph_id  = (const int*)  d_in[12];
  const int*   ncfg_graph_id= (const int*)  d_in[13];
  const float* emb          = (const float*)d_in[14];
  const float* pW0 = (const float*)d_in[15];
  const float* pb0 = (const float*)d_in[16];
  const float* pW1 = (const float*)d_in[17];
  const float* pb1 = (const float*)d_in[18];
  const float* gW0[2] = { (const float*)d_in[19], (const float*)d_in[23] };
  const float* gb0[2] = { (const float*)d_in[20], (const float*)d_in[24] };
  const float* gW1[2] = { (const float*)d_in[21], (const float*)d_in[25] };
  const float* gb1[2] = { (const float*)d_in[22], (const float*)d_in[26] };
  const float* postW0 = (const float*)d_in[27];
  const float* postW1 = (const float*)d_in[28];
  float* out = (float*)d_out;

  // ---- workspace layout (~180 MB) ----
  size_t off = 0;
  char* base = (char*)d_ws;
  auto take = [&](size_t elems) -> float* {
    float* r = (float*)(base + off);
    off += ((elems * sizeof(float)) + 255) & ~(size_t)255;
    return r;
  };
  float* op_full   = take((size_t)N_ * DOP_);
  float* dis       = take(N_);
  float* cfgs      = take((size_t)N_ * C_ * FCFG_);
  float* x_full    = take((size_t)N_ * CH_);
  float* x_bp      = take((size_t)N_ * CH_);
  float* agg       = take((size_t)N_ * C_ * DG_);
  float* cfg_nodes = take((size_t)NC_ * CH_);
  float* sum_op    = take((size_t)G_ * CH_);
  float* sum_cfg   = take((size_t)G_ * CH_);
  float* nops      = take(G_);
  float* pW0p      = take((size_t)D0P_ * H_);       // zero-padded prenet W0
  float* gW0p0     = take((size_t)DGP_ * H_);       // zero-padded gc0 W0
  float* gW0p1     = take((size_t)DGP_ * H_);       // zero-padded gc1 W0
  const float* gW0p[2] = { gW0p0, gW0p1 };

  // Zero-pad W0 matrices so the WMMA K-loops are branch-free.
  k_pad_w<<<blk(D0P_ * H_), 256, 0, stream>>>(pW0, pW0p, D0_, D0P_);
  k_pad_w<<<blk(DGP_ * H_), 256, 0, stream>>>(gW0[0], gW0p0, DG_, DGP_);
  k_pad_w<<<blk(DGP_ * H_), 256, 0, stream>>>(gW0[1], gW0p1, DG_, DGP_);

  // op_full = concat(op_feats, emb[op_codes])
  k_op_full<<<blk((long long)N_ * DOP_), 256, 0, stream>>>(op_feats, op_codes, emb, op_full);

  // full-graph pass and selected("bp") pass
  run_node_forward(feed_src, feed_tgt, config_src, config_tgt, config_feats,
                   op_full, dis, cfgs, agg, pW0p, pb0, pW1, pb1,
                   gW0p, gb0, gW1, gb1, x_full, stream);
  run_node_forward(s_feed_src, s_feed_tgt, s_config_src, s_config_tgt, config_feats,
                   op_full, dis, cfgs, agg, pW0p, pb0, pW1, pb1,
                   gW0p, gb0, gW1, gb1, x_bp, stream);

  // x = where(selected, x_bp, x_full)  (merge into x_full in place)
  k_select<<<blk((long long)N_ * CH_), 256, 0, stream>>>(selected, x_bp, x_full);

  // cfg_nodes = adj_config^T @ x
  k_fill<<<blk((long long)NC_ * CH_), 256, 0, stream>>>(cfg_nodes, (long long)NC_ * CH_, 0.f);
  k_cfg_nodes<<<blk((long long)NC_ * CH_), 256, 0, stream>>>(config_src, config_tgt, x_full, cfg_nodes);

  // pooling
  k_fill<<<1, 256, 0, stream>>>(nops, G_, 0.f);
  k_nops<<<blk(N_), 256, 0, stream>>>(op_graph_id, nops);
  k_fill<<<blk((long long)G_ * CH_), 256, 0, stream>>>(sum_op, (long long)G_ * CH_, 0.f);
  k_graph_sum<<<blk((long long)N_ * CH_), 256, 0, stream>>>(x_full, op_graph_id, sum_op,
                                                            (long long)N_ * CH_);
  k_fill<<<blk((long long)G_ * CH_), 256, 0, stream>>>(sum_cfg, (long long)G_ * CH_, 0.f);
  k_graph_sum<<<blk((long long)NC_ * CH_), 256, 0, stream>>>(cfg_nodes, ncfg_graph_id, sum_cfg,
                                                             (long long)NC_ * CH_);

  // postnet -> out [G, C]
  k_post<<<G_ * C_, 32, 0, stream>>>(sum_op, sum_cfg, nops, postW0, postW1, out);
}